// Attention_78950088835787
// MI455X (gfx1250) — compile-verified
//
#include <hip/hip_runtime.h>
#include <math.h>
#include <stdint.h>

// Problem constants (match reference)
#define BB 32
#define SS 2048
#define HH 1024
#define LL 4

// Tiling
#define CHUNK_ROWS 64              // S rows per workgroup
#define NCHUNK (SS / CHUNK_ROWS)   // 32 chunks per batch
#define NTILE (CHUNK_ROWS / 16)    // 4 WMMA row-tiles per chunk
#define NWAVE 8                    // 256 threads, wave32
#define WCOL 128                   // h-columns owned per wave (8 waves * 128 = 1024)

// LDS layout (floats). Row strides padded by 4 words for bank spread.
#define ENC_STRIDE 132                               // 128 + 4
#define ENC_FLOATS (NTILE * NWAVE * 16 * ENC_STRIDE) // 67584
#define DEC_STRIDE 1028                              // 1024 + 4
#define DEC_FLOATS (5 * DEC_STRIDE)                  // 5140 (row 4 = zeros)
#define SC_FLOATS (CHUNK_ROWS * 4)                   // 256
#define PT_STRIDE 68
#define PT_FLOATS (4 * PT_STRIDE)                    // 272
#define LDS_FLOATS (ENC_FLOATS + DEC_FLOATS + SC_FLOATS + PT_FLOATS + 4 + 4)
#define LDS_BYTES (LDS_FLOATS * 4)                   // 293,040 B < 320 KB

typedef __attribute__((ext_vector_type(2))) float v2f;
typedef __attribute__((ext_vector_type(8))) float v8f;

// Fused pass 1: per (batch, S-chunk) -> chunk-local softmax partials.
// enc: [B,S,H]  dec: [L,B,H]
// mPart,dPart: [B, NCHUNK, 4]   accPart: [B, NCHUNK, 4, H]
__global__ __launch_bounds__(256) void attn_partial_kernel(
    const float* __restrict__ enc, const float* __restrict__ dec,
    float* __restrict__ mPart, float* __restrict__ dPart,
    float* __restrict__ accPart) {
  extern __shared__ float smem[];
  float* encT    = smem;                       // [tile][wave][16][ENC_STRIDE]
  float* decT    = encT + ENC_FLOATS;          // [5][DEC_STRIDE]
  float* scoresS = decT + DEC_FLOATS;          // [64][4]
  float* PT      = scoresS + SC_FLOATS;        // [4][PT_STRIDE]  exp(score-m)
  float* mS      = PT + PT_FLOATS;             // [4]
  float* dS      = mS + 4;                     // [4]

  const int b  = blockIdx.x >> 5;              // batch
  const int c  = blockIdx.x & 31;              // chunk
  const int s0 = c * CHUNK_ROWS;

  const int wv    = threadIdx.x >> 5;          // wave 0..7
  const int ln    = threadIdx.x & 31;          // lane
  const int row16 = ln & 15;
  const int kOff  = (ln >> 4) << 1;            // 0 or 2 (A/B K sub-lane split)

  // ---- kick off the whole enc chunk as async global->LDS (no VGPR bounce) ----
  // Each wave issues 64 x B128 (16 rows x 4 tiles) back-to-back on ASYNCcnt,
  // giving ~32 KB in flight per wave / ~256 KB per WGP to cover HBM latency.
  {
    const unsigned encLdsBase =
        (unsigned)(uintptr_t)(void*)encT + (unsigned)(ln * 16);  // + lane's 16B
    const float* gBase = enc + ((size_t)(b * SS + s0)) * HH + wv * WCOL + ln * 4;
    #pragma unroll
    for (int tile = 0; tile < NTILE; ++tile) {
      #pragma unroll
      for (int r = 0; r < 16; ++r) {
        const unsigned dstOff =
            encLdsBase +
            (unsigned)(((tile * NWAVE + wv) * 16 + r) * ENC_STRIDE) * 4u;
        const float* g = gBase + (size_t)(tile * 16 + r) * HH;
        asm volatile("global_load_async_to_lds_b128 %0, %1, off"
                     :: "v"(dstOff), "v"(g)
                     : "memory");
      }
    }
  }

  // ---- stage decoder_hidden[l, b, :] (4x1024) + zero row 4 + zero scores ----
  {
    const int t = threadIdx.x;                 // 256 threads * 4 floats = 1024/row
    #pragma unroll
    for (int l = 0; l < 4; ++l) {
      const float4 v = *(const float4*)(dec + ((size_t)(l * BB + b)) * HH + t * 4);
      *(float4*)(decT + l * DEC_STRIDE + t * 4) = v;
    }
    *(float4*)(decT + 4 * DEC_STRIDE + t * 4) = make_float4(0.f, 0.f, 0.f, 0.f);
    scoresS[t] = 0.f;
  }
  __syncthreads();

  // ---- GEMM1: scores[s,l] = enc[s,:] . dec[l,:]   (split-K over waves) ----
  // A 16x4 (M=s, K=h): lane row16 holds K=kOff..kOff+1.  B 4x16 (K=h, N=l): rows>=4 read zero row.
  // Async loads complete in order per wave -> staged waits pipeline tile 0
  // compute under tiles 1..3 streaming.
  const int rsel = (row16 < 4) ? row16 : 4;
  #pragma unroll
  for (int tile = 0; tile < NTILE; ++tile) {
    if (tile == 0)      asm volatile("s_wait_asynccnt 0x30" ::: "memory");
    else if (tile == 1) asm volatile("s_wait_asynccnt 0x20" ::: "memory");
    else if (tile == 2) asm volatile("s_wait_asynccnt 0x10" ::: "memory");
    else                asm volatile("s_wait_asynccnt 0x0"  ::: "memory");

    v8f acc = {};
    const float* eT = encT + (size_t)((tile * NWAVE + wv) * 16 + row16) * ENC_STRIDE;
    const float* dT = decT + rsel * DEC_STRIDE + wv * WCOL;
    #pragma unroll 8
    for (int k = 0; k < WCOL; k += 4) {
      v2f a  = *(const v2f*)(eT + k + kOff);
      v2f bb = *(const v2f*)(dT + k + kOff);
      acc = __builtin_amdgcn_wmma_f32_16x16x4_f32(false, a, false, bb,
                                                  (short)0, acc, false, false);
    }
    // D: VGPR r, lanes 0-15 -> [M=r][N=ln], lanes 16-31 -> [M=r+8][N=ln-16]; N<4 valid.
    if ((ln & 15) < 4) {
      const int l = ln & 15;
      const int sBase = tile * 16 + ((ln < 16) ? 0 : 8);
      #pragma unroll
      for (int r = 0; r < 8; ++r) atomicAdd(&scoresS[(sBase + r) * 4 + l], acc[r]);
    }
  }
  __syncthreads();

  // ---- chunk-local stable softmax over the 64 rows ----
  if (threadIdx.x < 4) {
    const int l = threadIdx.x;
    float mx = -INFINITY;
    for (int s = 0; s < CHUNK_ROWS; ++s) mx = fmaxf(mx, scoresS[s * 4 + l]);
    mS[l] = mx;
    dS[l] = 0.f;
  }
  __syncthreads();
  {
    const int s = threadIdx.x >> 2, l = threadIdx.x & 3;
    const float p = __expf(scoresS[s * 4 + l] - mS[l]);
    PT[l * PT_STRIDE + s] = p;
    atomicAdd(&dS[l], p);
  }
  __syncthreads();
  if (threadIdx.x < 4) {
    const int l = threadIdx.x;
    mPart[(b * NCHUNK + c) * 4 + l] = mS[l];
    dPart[(b * NCHUNK + c) * 4 + l] = dS[l];
  }

  // ---- GEMM2: acc[l,h] += P[l,s] * enc[s,h]  (M=l pad16, N=h, K=s) ----
  v8f acc2[8];
  #pragma unroll
  for (int nt = 0; nt < 8; ++nt) acc2[nt] = (v8f){};
  const int n15 = ln & 15;
  for (int tile = 0; tile < NTILE; ++tile) {
    const float* eB = encT + (size_t)((tile * NWAVE + wv) * 16) * ENC_STRIDE;
    #pragma unroll
    for (int ks = 0; ks < 4; ++ks) {
      // A rows 4..15 duplicate rows 0..3 (their D rows are never read)
      v2f a2 = *(const v2f*)(PT + (row16 & 3) * PT_STRIDE + tile * 16 + ks * 4 + kOff);
      const int kk = ks * 4 + kOff;
      #pragma unroll
      for (int nt = 0; nt < 8; ++nt) {
        v2f b2;
        b2.x = eB[(kk + 0) * ENC_STRIDE + nt * 16 + n15];
        b2.y = eB[(kk + 1) * ENC_STRIDE + nt * 16 + n15];
        acc2[nt] = __builtin_amdgcn_wmma_f32_16x16x4_f32(false, a2, false, b2,
                                                         (short)0, acc2[nt], false, false);
      }
    }
  }

  // ---- write partial context: rows M=0..3 (lanes 0-15) are the 4 l's ----
  if (ln < 16) {
    float* op = accPart + ((size_t)(b * NCHUNK + c) * 4) * HH + wv * WCOL + ln;
    #pragma unroll
    for (int nt = 0; nt < 8; ++nt)
      #pragma unroll
      for (int r = 0; r < 4; ++r)
        op[(size_t)r * HH + nt * 16] = acc2[nt][r];
  }
}

// Pass 2: merge chunk partials with log-sum-exp rescale. out: [L,B,H]
__global__ __launch_bounds__(256) void attn_combine_kernel(
    const float* __restrict__ mPart, const float* __restrict__ dPart,
    const float* __restrict__ accPart, float* __restrict__ out) {
  const int idx = blockIdx.x * 256 + threadIdx.x;   // over L*B*H = 131072
  const int h = idx & (HH - 1);
  const int b = (idx >> 10) & (BB - 1);
  const int l = idx >> 15;
  float M = -INFINITY;
  #pragma unroll 4
  for (int c = 0; c < NCHUNK; ++c)
    M = fmaxf(M, mPart[(b * NCHUNK + c) * 4 + l]);
  float D = 0.f, A = 0.f;
  #pragma unroll 4
  for (int c = 0; c < NCHUNK; ++c) {
    const float w = __expf(mPart[(b * NCHUNK + c) * 4 + l] - M);
    D += w * dPart[(b * NCHUNK + c) * 4 + l];
    A += w * accPart[((size_t)(b * NCHUNK + c) * 4 + l) * HH + h];
  }
  out[idx] = A / D;
}

extern "C" void kernel_launch(void* const* d_in, const int* in_sizes, int n_in,
                              void* d_out, int out_size, void* d_ws, size_t ws_size,
                              hipStream_t stream) {
  (void)in_sizes; (void)n_in; (void)out_size; (void)ws_size;
  const float* enc = (const float*)d_in[0];   // [B,S,H] f32
  const float* dec = (const float*)d_in[1];   // [L,B,H] f32
  float* out = (float*)d_out;                 // [L,B,H] f32

  // Workspace carve: m (4K f) | d (4K f) | acc (4M f) = ~16.03 MB
  float* mPart = (float*)d_ws;
  float* dPart = mPart + BB * NCHUNK * 4;
  float* accPart = dPart + BB * NCHUNK * 4;

  // Allow >64KB dynamic LDS (293,040 B; WGP supports 320 KB).
  hipFuncSetAttribute((const void*)attn_partial_kernel,
                      hipFuncAttributeMaxDynamicSharedMemorySize, LDS_BYTES);

  attn_partial_kernel<<<dim3(BB * NCHUNK), dim3(256), LDS_BYTES, stream>>>(
      enc, dec, mPart, dPart, accPart);
  attn_combine_kernel<<<dim3((LL * BB * HH) / 256), dim3(256), 0, stream>>>(
      mPart, dPart, accPart, out);
}